// CotLaplacian_39814346834441
// MI455X (gfx1250) — compile-verified
//
#include <hip/hip_runtime.h>

// Problem constants from the reference: B=8, N=200000, FN=400000
#define B_  8
#define N_  200000
#define FN_ 400000

// ---------------------------------------------------------------------------
// Kernel 1 (listed first so its asm shows in the snippet): one thread per face.
//   Gather v1,v2,v3; Heron-area cot weights exactly as the reference;
//   fused scatter: out[i] += sum over incident edges of w*(x[j]-x[i]),
//   which algebraically absorbs the deg[i]*x[i] term -> single pass,
//   9 returnless fp32 global atomics per face (vs 24 naive).
// ---------------------------------------------------------------------------
__global__ __launch_bounds__(256) void CotLap_scatter_kernel(
    const float* __restrict__ V,   // (B, N, 3)
    const int*   __restrict__ F,   // (B, FN, 3)
    float*       __restrict__ out) // (B*N, 3)
{
    int t = blockIdx.x * 256 + threadIdx.x;            // face id in [0, B*FN)
    const int total = B_ * FN_;
    if (t >= total) return;

    int b  = t / FN_;
    int fo = t * 3;

    // Prefetch the F stream ~24 KB ahead (speculative; OOB silently dropped).
    __builtin_prefetch(F + fo + 2048 * 3, 0, 0);

    int i1 = F[fo + 0];
    int i2 = F[fo + 1];
    int i3 = F[fo + 2];

    const float* vb = V + (size_t)b * (N_ * 3);
    const float* q1 = vb + (size_t)i1 * 3;
    const float* q2 = vb + (size_t)i2 * 3;
    const float* q3 = vb + (size_t)i3 * 3;

    float p1x = q1[0], p1y = q1[1], p1z = q1[2];
    float p2x = q2[0], p2y = q2[1], p2z = q2[2];
    float p3x = q3[0], p3y = q3[1], p3z = q3[2];

    // Squared edge lengths (l1 opposite v1, etc.), matching the reference.
    float dx, dy, dz;
    dx = p2x - p3x; dy = p2y - p3y; dz = p2z - p3z;
    float l1s = dx*dx + dy*dy + dz*dz;
    dx = p3x - p1x; dy = p3y - p1y; dz = p3z - p1z;
    float l2s = dx*dx + dy*dy + dz*dz;
    dx = p1x - p2x; dy = p1y - p2y; dz = p1z - p2z;
    float l3s = dx*dx + dy*dy + dz*dz;

    float l1 = sqrtf(l1s);
    float l2 = sqrtf(l2s);
    float l3 = sqrtf(l3s);

    float sp = 0.5f * (l1 + l2 + l3);
    float A  = 2.0f * sqrtf(sp * (sp - l1) * (sp - l2) * (sp - l3));
    float inv = 1.0f / (4.0f * A);

    float w0 = (l2s + l3s - l1s) * inv;   // cot23 : edge (v2, v3)
    float w1 = (l1s + l3s - l2s) * inv;   // cot31 : edge (v3, v1)
    float w2 = (l1s + l2s - l3s) * inv;   // cot12 : edge (v1, v2)

    // Per-vertex fused contributions: out[i] += sum_e w_e * (x[j] - x[i])
    float g1x = w1 * (p3x - p1x) + w2 * (p2x - p1x);
    float g1y = w1 * (p3y - p1y) + w2 * (p2y - p1y);
    float g1z = w1 * (p3z - p1z) + w2 * (p2z - p1z);

    float g2x = w0 * (p3x - p2x) + w2 * (p1x - p2x);
    float g2y = w0 * (p3y - p2y) + w2 * (p1y - p2y);
    float g2z = w0 * (p3z - p2z) + w2 * (p1z - p2z);

    float g3x = w0 * (p2x - p3x) + w1 * (p1x - p3x);
    float g3y = w0 * (p2y - p3y) + w1 * (p1y - p3y);
    float g3z = w0 * (p2z - p3z) + w1 * (p1z - p3z);

    int base = b * N_;
    float* o1 = out + (size_t)(base + i1) * 3;
    float* o2 = out + (size_t)(base + i2) * 3;
    float* o3 = out + (size_t)(base + i3) * 3;

    // Native returnless fp32 global atomics (global_atomic_add_f32).
    unsafeAtomicAdd(o1 + 0, g1x);
    unsafeAtomicAdd(o1 + 1, g1y);
    unsafeAtomicAdd(o1 + 2, g1z);
    unsafeAtomicAdd(o2 + 0, g2x);
    unsafeAtomicAdd(o2 + 1, g2y);
    unsafeAtomicAdd(o2 + 2, g2z);
    unsafeAtomicAdd(o3 + 0, g3x);
    unsafeAtomicAdd(o3 + 1, g3y);
    unsafeAtomicAdd(o3 + 2, g3z);
}

// ---------------------------------------------------------------------------
// Kernel 2: zero d_out (harness poisons it with 0xAA before timing).
// Single guarded float4 store per thread; tail (n % 4, never taken for
// n = 4,800,000) handled by one thread to keep codegen compact.
// ---------------------------------------------------------------------------
__global__ __launch_bounds__(256) void CotLap_zero_kernel(float* __restrict__ out, int n) {
    int i  = blockIdx.x * 256 + threadIdx.x;
    int n4 = n >> 2;
    if (i < n4) {
        reinterpret_cast<float4*>(out)[i] = make_float4(0.f, 0.f, 0.f, 0.f);
    }
    if (i == 0) {
        for (int k = n & ~3; k < n; ++k) out[k] = 0.f;
    }
}

extern "C" void kernel_launch(void* const* d_in, const int* in_sizes, int n_in,
                              void* d_out, int out_size, void* d_ws, size_t ws_size,
                              hipStream_t stream) {
    const float* V = (const float*)d_in[0];   // (B, N, 3) fp32
    const int*   F = (const int*)d_in[1];     // (B, FN, 3) int32
    float* out = (float*)d_out;               // (B*N, 3) fp32

    // 1) zero the output accumulator
    int n  = out_size;                        // B*N*3 = 4,800,000
    int n4 = n >> 2;
    CotLap_zero_kernel<<<(n4 + 255) / 256, 256, 0, stream>>>(out, n);

    // 2) fused cot-weight + symmetric-Laplacian scatter
    const int total  = B_ * FN_;              // 3,200,000 faces
    const int blocks = (total + 255) / 256;   // 12,500 blocks of 8 wave32s
    CotLap_scatter_kernel<<<blocks, 256, 0, stream>>>(V, F, out);
}